// GPKANLayer_59416577572914
// MI455X (gfx1250) — compile-verified
//
#include <hip/hip_runtime.h>
#include <hip/hip_bf16.h>
#include <math.h>

typedef __attribute__((ext_vector_type(2))) float v2f;
typedef __attribute__((ext_vector_type(8))) float v8f;

#define O_DIM 128
#define I_DIM 128
#define M_DIM 32
#define N_DIM 128
#define LOG2E 1.44269504088896340736f

// Per-wave LDS region layout (float indices)
#define SZ_OFF   0    // z[32]
#define SQ_OFF   32   // qs[32]   (qmu * sqrt(l2/den1))
#define SW_OFF   64   // w2s[32]  ((qvar+qmu^2) * sqrt(l2/den2))
#define CC_OFF   96   // c2sv replicated x32
#define PAD_OFF  128  // 32 zeros
#define WREG     160  // floats per wave region

// One block per o (128 blocks), 8 waves; wave w handles n-tile [16w, 16w+16).
// Two WMMA f32 16x16x4 chains per (o,i): A = weight rows (loaded from LDS with
// the row-masking folded into the per-lane base address), B = exp kernel values.
__global__ __launch_bounds__(256)
void gpkan_moment_kernel(const float* __restrict__ x,
                         const float* __restrict__ z,
                         const float* __restrict__ qmu,
                         const float* __restrict__ qlv,
                         const float* __restrict__ lsc,
                         const float* __restrict__ lva,
                         float* __restrict__ out)
{
    __shared__ __attribute__((aligned(16))) float smem[8 * WREG];

    const int o    = blockIdx.x;
    const int wave = threadIdx.x >> 5;   // 0..7 -> n tile
    const int lane = threadIdx.x & 31;
    const int row  = lane & 15;          // A-matrix row / result sample index
    const int half = lane >> 4;          // selects K slots {0,1} vs {2,3}
    const int n    = wave * 16 + row;

    float* W = &smem[wave * WREG];

    // zero pad region once (each wave owns its region; no cross-wave sync needed)
    W[PAD_OFF + lane] = 0.0f;

    // per-lane base offsets into the wave region (row masking folded into address)
    const int h2 = 2 * half;
    const int a1off = ((row == 0) ? SQ_OFF : PAD_OFF) + h2;
    const int a2off = ((row == 0) ? SW_OFF : ((row == 1) ? CC_OFF : PAD_OFF)) + h2;
    const int zoff  = SZ_OFF + h2;

    float ym = 0.f, yv = 0.f;

    #pragma unroll 1
    for (int i = 0; i < I_DIM; ++i) {
        const int oi = o * I_DIM + i;

        // per-(o,i) scalars (lane-uniform)
        float scale = fmaxf(__expf(lsc[oi]), 0.1f);
        float l2    = scale * scale;
        float sv    = fmaxf(__expf(lva[oi]), 1e-5f);
        float den1  = l2 + 1e-6f;        // l2 + x_var
        float den2  = l2 + 2e-6f;        // l2 + 2*x_var
        float r1    = sqrtf(l2 / den1);  // c1/sv
        float r2    = sqrtf(l2 / den2);  // c2/sv^2
        float c2sv  = sv * r2;           // c2/sv
        float k1    = -0.5f * LOG2E / den1;
        float k2    = -LOG2E / den2;

        // per-lane inducing point data: m = lane (wave32 => m in 0..31 exactly)
        const float* zp  = z   + (size_t)oi * M_DIM;
        const float* qp  = qmu + (size_t)oi * M_DIM;
        const float* qvp = qlv + (size_t)oi * M_DIM;
        float z_l  = zp[lane];
        float qm_l = qp[lane];
        float qv_l = fmaxf(__expf(qvp[lane]), 1e-5f);

        float xn = x[n * I_DIM + i];

        // stage this i's broadcast data (prior iteration's DS loads are in-order
        // ahead of these stores; wave_barrier stops compiler reordering)
        __builtin_amdgcn_wave_barrier();
        W[SZ_OFF + lane] = z_l;
        W[SQ_OFF + lane] = qm_l * r1;                  // mean-row weight
        W[SW_OFF + lane] = (qv_l + qm_l * qm_l) * r2;  // second-moment-row weight
        W[CC_OFF + lane] = c2sv;                       // row-1 constant, replicated
        __builtin_amdgcn_wave_barrier();

        v8f acc1 = {};
        v8f acc2 = {};

        #pragma unroll
        for (int c = 0; c < 8; ++c) {
            // this lane's two K-slots map to m = 4c + 2*half + {0,1}
            v2f zp2 = *(const v2f*)(W + zoff  + 4 * c);  // ds_load_b64
            v2f a1  = *(const v2f*)(W + a1off + 4 * c);  // ds_load_b64
            v2f a2  = *(const v2f*)(W + a2off + 4 * c);  // ds_load_b64

            float d0  = xn - zp2[0];
            float d1  = xn - zp2[1];
            float d0s = d0 * d0;
            float d1s = d1 * d1;
            // B matrices: e-values, B[k, col=n] -> lane holds col = lane&15
            v2f b1 = { __builtin_amdgcn_exp2f(d0s * k1),
                       __builtin_amdgcn_exp2f(d1s * k1) };
            v2f b2 = { __builtin_amdgcn_exp2f(d0s * k2),
                       __builtin_amdgcn_exp2f(d1s * k2) };

            acc1 = __builtin_amdgcn_wmma_f32_16x16x4_f32(
                       false, a1, false, b1, (short)0, acc1, false, false);
            acc2 = __builtin_amdgcn_wmma_f32_16x16x4_f32(
                       false, a2, false, b2, (short)0, acc2, false, false);
        }

        // D row0/row1 live in VGPR 0/1, lanes 0..15, col = lane = n-offset.
        float em    = acc1[0];
        float secnd = acc2[0];
        float s2b   = acc2[1];
        float ev = sv - s2b + secnd - em * em;
        ev = fmaxf(ev, 1e-6f);
        ym += em;
        yv += ev;
    }

    if (lane < 16) {
        out[n * O_DIM + o]                 = ym;  // y_mean [N,O]
        out[N_DIM * O_DIM + n * O_DIM + o] = yv;  // y_var  [N,O]
    }
}

extern "C" void kernel_launch(void* const* d_in, const int* in_sizes, int n_in,
                              void* d_out, int out_size, void* d_ws, size_t ws_size,
                              hipStream_t stream) {
    const float* x   = (const float*)d_in[0];  // [N, I]
    const float* z   = (const float*)d_in[1];  // [O, I, M]
    const float* qmu = (const float*)d_in[2];  // [O, I, M]
    const float* qlv = (const float*)d_in[3];  // [O, I, M]
    const float* lsc = (const float*)d_in[4];  // [O, I]
    const float* lva = (const float*)d_in[5];  // [O, I]
    float* out = (float*)d_out;                // y_mean ++ y_var, 2*N*O floats

    dim3 grid(O_DIM);
    dim3 block(256);
    gpkan_moment_kernel<<<grid, block, 0, stream>>>(x, z, qmu, qlv, lsc, lva, out);
}